// MiniBatch1d_42279658062013
// MI455X (gfx1250) — compile-verified
//
#include <hip/hip_runtime.h>
#include <stdint.h>

typedef float v2f __attribute__((ext_vector_type(2)));
typedef float v8f __attribute__((ext_vector_type(8)));

#define IN_F    1024
#define OUT_F   64
#define INNER   16
#define BATCH   512
#define OUT_COLS (IN_F + OUT_F)   // 1088

// ---------------------------------------------------------------------------
// Kernel 0: Wt[n][k] = W[k][n]  (1024x1024 f32, tiled LDS transpose).
// Makes the WMMA B-operand fetches K-contiguous (b64 instead of 2 strided b32).
// ---------------------------------------------------------------------------
__global__ __launch_bounds__(256) void transpose_w(const float* __restrict__ W,
                                                   float* __restrict__ Wt) {
  __shared__ float tile[32][33];            // +1 pad: no bank conflicts
  const int bx = blockIdx.x * 32;           // N base
  const int by = blockIdx.y * 32;           // K base
  const int tx = threadIdx.x;               // 0..31
  const int ty = threadIdx.y;               // 0..7
  #pragma unroll
  for (int r = ty; r < 32; r += 8)
    tile[r][tx] = W[(by + r) * IN_F + bx + tx];
  __syncthreads();
  #pragma unroll
  for (int r = ty; r < 32; r += 8)
    Wt[(bx + r) * IN_F + by + tx] = tile[tx][r];
}

// ---------------------------------------------------------------------------
// Kernel 1: m[512,1024] = x @ W via V_WMMA_F32_16X16X4_F32.
// One wave32 computes a 16(M) x 64(N) strip: 4 v8f accumulators, so each
// A-load (b64) feeds 4 WMMAs; B-loads are contiguous b64 from Wt.
// 512 waves total -> 128 blocks of 4 waves.
// ---------------------------------------------------------------------------
__global__ __launch_bounds__(128) void gemm_wmma_f32(const float* __restrict__ x,
                                                     const float* __restrict__ Wt,
                                                     float* __restrict__ m) {
  const int wave  = threadIdx.x >> 5;
  const int lane  = threadIdx.x & 31;
  const int gid   = blockIdx.x * 4 + wave;
  const int tileM = gid >> 4;               // 0..31  (512/16 row tiles)
  const int nquad = gid & 15;               // 0..15  (1024/64 col strips)

  const int mrow  = lane & 15;              // M (A) or N (B/C/D) within tile
  const int khalf = (lane >> 4) * 2;        // lanes 0-15 -> K{0,1}, 16-31 -> K{2,3}
  const int row   = tileM * 16 + mrow;

  const float* __restrict__ xrow = x + row * IN_F + khalf;
  const float* __restrict__ wt0  = Wt + (nquad * 64 + mrow) * IN_F + khalf;

  v8f acc0 = {0.f,0.f,0.f,0.f,0.f,0.f,0.f,0.f};
  v8f acc1 = acc0, acc2 = acc0, acc3 = acc0;

  #pragma unroll 2
  for (int k = 0; k < IN_F; k += 4) {
    v2f a = *(const v2f*)(xrow + k);        // A: row mrow, K = k+khalf, k+khalf+1
    v2f b0 = *(const v2f*)(wt0 + k);                 // B tile n0, col mrow
    v2f b1 = *(const v2f*)(wt0 + 16 * IN_F + k);     // B tile n0+16
    v2f b2 = *(const v2f*)(wt0 + 32 * IN_F + k);     // B tile n0+32
    v2f b3 = *(const v2f*)(wt0 + 48 * IN_F + k);     // B tile n0+48
    acc0 = __builtin_amdgcn_wmma_f32_16x16x4_f32(false, a, false, b0, (short)0, acc0, false, false);
    acc1 = __builtin_amdgcn_wmma_f32_16x16x4_f32(false, a, false, b1, (short)0, acc1, false, false);
    acc2 = __builtin_amdgcn_wmma_f32_16x16x4_f32(false, a, false, b2, (short)0, acc2, false, false);
    acc3 = __builtin_amdgcn_wmma_f32_16x16x4_f32(false, a, false, b3, (short)0, acc3, false, false);
  }

  // C/D layout: VGPR g holds M = g + 8*(lane>>4), N = lane&15
  const int mbase = tileM * 16 + (lane >> 4) * 8;
  float* __restrict__ mout = m + mbase * IN_F + nquad * 64 + mrow;
  #pragma unroll
  for (int g = 0; g < 8; ++g) {
    mout[g * IN_F +  0] = acc0[g];
    mout[g * IN_F + 16] = acc1[g];
    mout[g * IN_F + 32] = acc2[g];
    mout[g * IN_F + 48] = acc3[g];
  }
}

// ---------------------------------------------------------------------------
// Kernel 2: o[i,f] = mean_j exp(-sum_k |m[i,f,k] - m[j,f,k]|) -> out[:,1024+f].
// The 32 KB m[:, f, :] slab is staged into LDS with the CDNA5 async path
// (global_load_async_to_lds_b128, ASYNCcnt-tracked, no VGPR round-trip).
// ---------------------------------------------------------------------------
__global__ __launch_bounds__(256) void pairwise_l1_exp(const float* __restrict__ m,
                                                       float* __restrict__ out) {
  __shared__ float sm[BATCH * INNER];   // 512*16 f32 = 32 KB (of 320 KB WGP LDS)
  const int tid = threadIdx.x;
  const int f   = blockIdx.y;

  // Low 32 bits of the generic address of a __shared__ object are the LDS
  // byte offset (aperture rule, ISA 10.2) — exactly what async VDST wants.
  const uint32_t lds_base = (uint32_t)(uintptr_t)(&sm[0]);
  const float* gbase = m + f * INNER;   // m[j, f, k] = m[j*1024 + f*16 + k]
  #pragma unroll
  for (int idx = tid; idx < BATCH * 4; idx += 256) {
    const int j  = idx >> 2;            // batch row
    const int kq = idx & 3;             // which float4 of the 16
    const float*   gaddr = gbase + j * IN_F + kq * 4;
    const uint32_t laddr = lds_base + (uint32_t)idx * 16u;
    asm volatile("global_load_async_to_lds_b128 %0, %1, off"
                 :: "v"(laddr), "v"(gaddr) : "memory");
  }
  asm volatile("s_wait_asynccnt 0x0" ::: "memory");
  __syncthreads();

  const int i = blockIdx.x * 256 + tid;
  float mi[INNER];
  #pragma unroll
  for (int k = 0; k < INNER; ++k) mi[k] = sm[i * INNER + k];

  float acc = 0.f;
  for (int j = 0; j < BATCH; ++j) {
    float d = 0.f;
    #pragma unroll
    for (int k = 0; k < INNER; ++k) {
      d += __builtin_fabsf(mi[k] - sm[j * INNER + k]);   // broadcast LDS read
    }
    acc += __expf(-d);
  }
  out[i * OUT_COLS + IN_F + f] = acc * (1.0f / (float)BATCH);
}

// ---------------------------------------------------------------------------
// Kernel 3: out[:, 0:1024] = x (float4 copy; 1088-float stride keeps 16B align).
// ---------------------------------------------------------------------------
__global__ __launch_bounds__(256) void copy_x(const float* __restrict__ x,
                                              float* __restrict__ out) {
  const int idx = blockIdx.x * 256 + threadIdx.x;  // float4 index, 512*256 total
  const int i   = idx >> 8;
  const int c   = (idx & 255) * 4;
  *(float4*)(out + i * OUT_COLS + c) = *(const float4*)(x + i * IN_F + c);
}

extern "C" void kernel_launch(void* const* d_in, const int* in_sizes, int n_in,
                              void* d_out, int out_size, void* d_ws, size_t ws_size,
                              hipStream_t stream) {
  const float* x = (const float*)d_in[0];          // [512, 1024]
  const float* W = (const float*)d_in[1];          // T viewed as [1024(K), 1024(N)]
  float* out = (float*)d_out;                      // [512, 1088]

  float* Wt = (float*)d_ws;                        // [1024, 1024] transposed W (4 MB)
  float* m  = (float*)((char*)d_ws + (size_t)IN_F * IN_F * sizeof(float)); // 2 MB

  transpose_w<<<dim3(32, 32), dim3(32, 8), 0, stream>>>(W, Wt);
  gemm_wmma_f32<<<128, 128, 0, stream>>>(x, Wt, m);
  copy_x<<<512, 256, 0, stream>>>(x, out);
  pairwise_l1_exp<<<dim3(2, 64), 256, 0, stream>>>(m, out);
}